// GIPEPS_11252814315971
// MI455X (gfx1250) — compile-verified
//
#include <hip/hip_runtime.h>

// GI-PEPS amplitude, 6x6, N=2, DMAX=4. Meet-in-the-middle exact contraction:
//   top boundary <- rows 0,1,2 ; bottom boundary <- rows 5,4,3 ;
//   amplitude = MPS-MPS overlap via (r_top, r_bot) transfer vector.
// All contractions are GEMMs with K a multiple of 4 -> V_WMMA_F32_16X16X4_F32.
// Site slices are staged to LDS with GLOBAL_LOAD_ASYNC_TO_LDS_B32 (ASYNCcnt).

typedef float v2f __attribute__((ext_vector_type(2)));
typedef float v8f __attribute__((ext_vector_type(8)));

__device__ __forceinline__ v8f wmma4(v2f a, v2f b, v8f c) {
  // D = A(16x4) * B(4x16) + C  (f32 in/out, K=4)
  return __builtin_amdgcn_wmma_f32_16x16x4_f32(false, a, false, b, (short)0, c,
                                               false, false);
}

struct RowPtrs { const float* t[6]; };

#define SLAB 16384  // max boundary tensor per column after 3 rows: 64*4*64 f32

// One block per column c. Absorb site (r,c) into the boundary MPS tensor.
//   top:    T(l,x,d,r,y) = sum_u  B(l,u,r) * A(u,d,x,y)
//   bottom: T(l,x,u,r,y) = sum_d  B(l,d,r) * A(u,d,x,y)
// GEMM view: out[(l,rr), (mid,x,y)] = sum_k B[(l,rr),k] * A'[k,(mid,x,y)], K<=4.
__global__ __launch_bounds__(256) void absorb_kernel(RowPtrs tp, const int* __restrict__ sample,
                                                     const float* __restrict__ bndIn,
                                                     float* __restrict__ bndOut,
                                                     int r, int top, int first) {
  const int c = blockIdx.x;
  const int tid = threadIdx.x;
  const int lane = tid & 31, wv = tid >> 5;
  const int lo = lane & 15, hi = lane >> 4;

  bndIn  += c * SLAB;
  bndOut += c * SLAB;
  __builtin_prefetch(bndIn, 0, 1);  // global_prefetch_b8

  const int U  = (r > 0) ? 4 : 1;
  const int Dd = (r < 5) ? 4 : 1;
  const int X  = (c > 0) ? 4 : 1;
  const int Y  = (c < 5) ? 4 : 1;
  const int nprev = top ? r : (5 - r);          // rows already absorbed
  const int Lin = (c > 0) ? (1 << (2 * nprev)) : 1;
  const int Rin = (c < 5) ? (1 << (2 * nprev)) : 1;
  const int K      = top ? U : Dd;              // contracted virtual bond
  const int midDim = top ? Dd : U;              // surviving vertical bond
  const int Mm = Lin * Rin;
  const int Nn = midDim * X * Y;
  const int XY = X * Y;
  const int asz = U * Dd * X * Y;

  // ---- Gauss-law configuration index (matches reference _cfg_idx) ----
  const bool aL = c > 0, aR = c < 5, aU = r > 0, aD = r < 5;
  const int dep = aR ? 3 : (aD ? 2 : (aU ? 1 : 0));  // 0=L,1=U,2=D,3=R
  const int kL = aL ? sample[36 + r * 5 + (c - 1)] : 0;
  const int kR = aR ? sample[36 + r * 5 + c] : 0;
  const int kU = aU ? sample[66 + (r - 1) * 6 + c] : 0;
  const int kD = aD ? sample[66 + r * 6 + c] : 0;
  int idx = 0;
  if (aL && dep != 0) idx = idx * 2 + kL;
  if (aU && dep != 1) idx = idx * 2 + kU;
  if (aD && dep != 2) idx = idx * 2 + kD;
  if (aR && dep != 3) idx = idx * 2 + kR;
  const int nlinks = (int)aL + (int)aR + (int)aU + (int)aD;
  const int Nc = 1 << (nlinks - 1);             // nlinks >= 2 on a 6x6 grid
  const int p = sample[r * 6 + c];

  // ---- stage the gathered site slice A(u,d,x,y) into LDS (<=256 f32)
  //      via CDNA5 async global->LDS copy (tracked with ASYNCcnt) ----
  __shared__ float As[256];
  const float* Aslice = tp.t[c] + (size_t)(p * Nc + idx) * asz;
  if (tid < asz) {
    unsigned ldsAddr = (unsigned)(unsigned long long)&As[tid];  // addr[31:0] = LDS offset
    const float* gaddr = Aslice + tid;
    asm volatile("global_load_async_to_lds_b32 %0, %1, off"
                 :: "v"(ldsAddr), "v"(gaddr) : "memory");
  }
  asm volatile("s_wait_asynccnt 0x0" ::: "memory");
  __syncthreads();

  // ---- tiled GEMM, one K=4 WMMA per 16x16 output tile ----
  const int Mt = (Mm + 15) >> 4, Nt = (Nn + 15) >> 4;
  for (int t = wv; t < Mt * Nt; t += 8) {
    const int tm = t / Nt, tn = t % Nt;
    const int k0 = 2 * hi;

    v2f a;  // A-frag: boundary  B[(l,rr), k]
    {
      const int m = tm * 16 + lo;
      float a0 = 0.f, a1 = 0.f;
      if (m < Mm) {
        const int l = m / Rin, rr = m % Rin;
        if (k0 < K)     a0 = first ? 1.0f : bndIn[(l * K + k0) * Rin + rr];
        if (k0 + 1 < K) a1 = first ? 1.0f : bndIn[(l * K + k0 + 1) * Rin + rr];
      }
      a.x = a0; a.y = a1;
    }
    v2f b;  // B-frag: site  A'[k, (mid,x,y)]
    {
      const int n = tn * 16 + lo;
      float b0 = 0.f, b1 = 0.f;
      if (n < Nn) {
        const int mid = n / XY, rem = n % XY, x = rem / Y, y = rem % Y;
        const int base = x * Y + y;
        if (k0 < K) {
          const int uu = top ? k0 : mid, dd = top ? mid : k0;
          b0 = As[(uu * Dd + dd) * XY + base];
        }
        if (k0 + 1 < K) {
          const int uu = top ? (k0 + 1) : mid, dd = top ? mid : (k0 + 1);
          b1 = As[(uu * Dd + dd) * XY + base];
        }
      }
      b.x = b0; b.y = b1;
    }

    v8f acc = {0.f, 0.f, 0.f, 0.f, 0.f, 0.f, 0.f, 0.f};
    acc = wmma4(a, b, acc);

    // scatter D tile -> new boundary, C-order (L*X, midDim, Rin*Y)
    const int n = tn * 16 + lo;
    const int mid = n / XY, rem = n % XY, x = rem / Y, y = rem % Y;
    const int nOut = (x * midDim + mid) * (Rin * Y) + y;  // n-dependent part
    if (tm * 16 + 16 <= Mm && tn * 16 + 16 <= Nn) {
      // fast path: whole tile in bounds -> unpredicated stores
#pragma unroll
      for (int vv = 0; vv < 8; ++vv) {
        const int m = tm * 16 + vv + 8 * hi;
        const int l = m / Rin, rr = m % Rin;
        bndOut[(l * X) * midDim * (Rin * Y) + rr * Y + nOut] = acc[vv];
      }
    } else {
#pragma unroll
      for (int vv = 0; vv < 8; ++vv) {
        const int m = tm * 16 + vv + 8 * hi;
        if (m < Mm && n < Nn) {
          const int l = m / Rin, rr = m % Rin;
          bndOut[(l * X) * midDim * (Rin * Y) + rr * Y + nOut] = acc[vv];
        }
      }
    }
  }
}

// Single block: left-to-right MPS-MPS overlap with transfer vector v(r1,r2).
//  GEMM1: w(l2, d*Rt+r1) = sum_l1 v(l1,l2) * T(l1,d,r1)        K = Lc
//  GEMM2: v'(r1, r2)     = sum_{l2,d} w(l2,d,r1) * Bo(l2,d,r2) K = 4*Lc
__global__ __launch_bounds__(256) void overlap_kernel(const float* __restrict__ top,
                                                      const float* __restrict__ bot,
                                                      float* __restrict__ wbuf,
                                                      float* __restrict__ out) {
  __shared__ float vA[4096];
  __shared__ float vB[4096];
  const int tid = threadIdx.x;
  const int lane = tid & 31, wv = tid >> 5;
  const int lo = lane & 15, hi = lane >> 4;

  if (tid == 0) vA[0] = 1.0f;
  __syncthreads();

  float* vin = vA;
  float* vout = vB;
  for (int c = 0; c < 6; ++c) {
    const int Lc = (c > 0) ? 64 : 1;
    const int Rt = (c < 5) ? 64 : 1;  // Rb == Rt
    const float* T  = top + c * SLAB;
    const float* Bo = bot + c * SLAB;

    {  // GEMM1: M=Lc, N=4*Rt, K=Lc   -> wbuf (global scratch)
      const int M = Lc, N = 4 * Rt, Kk = Lc;
      const int Mt = (M + 15) >> 4, Nt = (N + 15) >> 4;
      for (int t = wv; t < Mt * Nt; t += 8) {
        const int tm = t / Nt, tn = t % Nt;
        v8f acc = {0.f, 0.f, 0.f, 0.f, 0.f, 0.f, 0.f, 0.f};
        for (int kk = 0; kk < Kk; kk += 4) {
          const int ka = kk + 2 * hi;
          const int m = tm * 16 + lo, n = tn * 16 + lo;
          v2f a, b;
          a.x = (m < M && ka < Kk)     ? vin[ka * Lc + m]       : 0.f;
          a.y = (m < M && ka + 1 < Kk) ? vin[(ka + 1) * Lc + m] : 0.f;
          b.x = (n < N && ka < Kk)     ? T[ka * N + n]          : 0.f;
          b.y = (n < N && ka + 1 < Kk) ? T[(ka + 1) * N + n]    : 0.f;
          acc = wmma4(a, b, acc);
        }
        const int n = tn * 16 + lo;
        if (tm * 16 + 16 <= M && tn * 16 + 16 <= N) {
#pragma unroll
          for (int vv = 0; vv < 8; ++vv) {
            const int m = tm * 16 + vv + 8 * hi;
            wbuf[m * N + n] = acc[vv];
          }
        } else {
#pragma unroll
          for (int vv = 0; vv < 8; ++vv) {
            const int m = tm * 16 + vv + 8 * hi;
            if (m < M && n < N) wbuf[m * N + n] = acc[vv];
          }
        }
      }
    }
    __threadfence();
    __syncthreads();
    {  // GEMM2: M=Rt, N=Rt, K=4*Lc  -> vout (LDS)
      const int M = Rt, N = Rt, Kk = 4 * Lc;
      const int Mt = (M + 15) >> 4, Nt = (N + 15) >> 4;
      for (int t = wv; t < Mt * Nt; t += 8) {
        const int tm = t / Nt, tn = t % Nt;
        v8f acc = {0.f, 0.f, 0.f, 0.f, 0.f, 0.f, 0.f, 0.f};
        for (int kk = 0; kk < Kk; kk += 4) {
          const int ka = kk + 2 * hi;
          const int m = tm * 16 + lo, n = tn * 16 + lo;
          v2f a, b;
          a.x = (m < M && ka < Kk)     ? wbuf[ka * Rt + m]       : 0.f;
          a.y = (m < M && ka + 1 < Kk) ? wbuf[(ka + 1) * Rt + m] : 0.f;
          b.x = (n < N && ka < Kk)     ? Bo[ka * N + n]          : 0.f;
          b.y = (n < N && ka + 1 < Kk) ? Bo[(ka + 1) * N + n]    : 0.f;
          acc = wmma4(a, b, acc);
        }
        const int n = tn * 16 + lo;
        if (tm * 16 + 16 <= M && tn * 16 + 16 <= N) {
#pragma unroll
          for (int vv = 0; vv < 8; ++vv) {
            const int m = tm * 16 + vv + 8 * hi;
            vout[m * N + n] = acc[vv];
          }
        } else {
#pragma unroll
          for (int vv = 0; vv < 8; ++vv) {
            const int m = tm * 16 + vv + 8 * hi;
            if (m < M && n < N) vout[m * N + n] = acc[vv];
          }
        }
      }
    }
    __syncthreads();
    float* tsw = vin; vin = vout; vout = tsw;
  }
  if (tid == 0) out[0] = vin[0];
}

static inline int siteSize(int r, int c) {
  const int U = (r > 0) ? 4 : 1, D = (r < 5) ? 4 : 1;
  const int X = (c > 0) ? 4 : 1, Y = (c < 5) ? 4 : 1;
  const int nl = (r > 0) + (r < 5) + (c > 0) + (c < 5);
  return 2 * (1 << (nl - 1)) * U * D * X * Y;
}

extern "C" void kernel_launch(void* const* d_in, const int* in_sizes, int n_in,
                              void* d_out, int out_size, void* d_ws, size_t ws_size,
                              hipStream_t stream) {
  (void)in_sizes; (void)out_size; (void)ws_size;
  // Site tensor pointers: either 36 separate leaves + sample, or one flat blob.
  const float* tens[36];
  if (n_in >= 37) {
    for (int i = 0; i < 36; ++i) tens[i] = (const float*)d_in[i];
  } else {
    const float* base = (const float*)d_in[0];
    size_t off = 0;
    for (int r = 0; r < 6; ++r)
      for (int c = 0; c < 6; ++c) { tens[r * 6 + c] = base + off; off += (size_t)siteSize(r, c); }
  }
  const int* sample = (const int*)d_in[n_in - 1];

  float* ws = (float*)d_ws;   // 25 * SLAB * 4B = 1.6 MB, resident in L2
  float* T0 = ws;
  float* T1 = ws + 6 * SLAB;
  float* S0 = ws + 12 * SLAB;
  float* S1 = ws + 18 * SLAB;
  float* W  = ws + 24 * SLAB;

  RowPtrs rp;
  auto row = [&](int r) { for (int c = 0; c < 6; ++c) rp.t[c] = tens[r * 6 + c]; return rp; };

  // top boundary: rows 0,1,2 (ping-pong T0/T1, final in T0)
  absorb_kernel<<<6, 256, 0, stream>>>(row(0), sample, T0, T0, 0, 1, 1);
  absorb_kernel<<<6, 256, 0, stream>>>(row(1), sample, T0, T1, 1, 1, 0);
  absorb_kernel<<<6, 256, 0, stream>>>(row(2), sample, T1, T0, 2, 1, 0);
  // bottom boundary: rows 5,4,3 (final in S0)
  absorb_kernel<<<6, 256, 0, stream>>>(row(5), sample, S0, S0, 5, 0, 1);
  absorb_kernel<<<6, 256, 0, stream>>>(row(4), sample, S0, S1, 4, 0, 0);
  absorb_kernel<<<6, 256, 0, stream>>>(row(3), sample, S1, S0, 3, 0, 0);
  // overlap -> scalar amplitude
  overlap_kernel<<<1, 256, 0, stream>>>(T0, S0, W, (float*)d_out);
}